// InvariantPointAttention_23476291240019
// MI455X (gfx1250) — compile-verified
//
#include <hip/hip_runtime.h>
#include <math.h>

// Problem constants (B=1)
#define NS   768   // sequence length
#define NH   12    // heads
#define SQK  16
#define PQK  4
#define SV   16
#define PV   8
#define CM   384
#define CPD  128   // CP
#define COUT 384
#define FACT_DIM 2112  // H*SV + 4*H*PV + H*CP = 192 + 384 + 1536

typedef __attribute__((ext_vector_type(2))) float v2f;
typedef __attribute__((ext_vector_type(8))) float v8f;

__device__ __forceinline__ v8f wmma_f32(v2f a, v2f b, v8f c) {
  // V_WMMA_F32_16X16X4_F32 : D = A(16x4) x B(4x16) + C(16x16), fp32 throughout
  return __builtin_amdgcn_wmma_f32_16x16x4_f32(false, a, false, b, (short)0, c, false, false);
}

// -------------------------------------------------------------------------
// Generic fp32 WMMA GEMM: Y[M][Nc] = A[M][K] @ W[K][Nc] + bias[Nc]
// grid = (Nc/16, M/16), block = 32 (one wave per 16x16 tile)
// -------------------------------------------------------------------------
__global__ void gemm_wmma_kernel(const float* __restrict__ A, const float* __restrict__ W,
                                 const float* __restrict__ bias, float* __restrict__ Y,
                                 int M, int K, int Nc) {
  const int lane = threadIdx.x & 31;
  const int lh   = lane >> 4;      // which half of the wave (selects K pair)
  const int l16  = lane & 15;
  const int arow = blockIdx.y * 16 + l16;
  const int bcol = blockIdx.x * 16 + l16;
  v8f acc = {};
  for (int k = 0; k < K; k += 4) {
    const int ka = k + 2 * lh;
    v2f a, b;
    const float* ap = A + (size_t)arow * K + ka;
    a.x = ap[0]; a.y = ap[1];
    b.x = W[(size_t)ka * Nc + bcol];
    b.y = W[(size_t)(ka + 1) * Nc + bcol];
    acc = wmma_f32(a, b, acc);
  }
  const float bs = bias[bcol];
#pragma unroll
  for (int v = 0; v < 8; ++v) {
    const int r = blockIdx.y * 16 + v + 8 * lh;
    Y[(size_t)r * Nc + bcol] = acc[v] + bs;
  }
}

// -------------------------------------------------------------------------
// Apply per-residue rotation/translation to point projections; repack so the
// 12 (q/k) or 24 (v) point dims are contiguous per (n,h)  -> WMMA-friendly.
// grid = NS, block = 192 (48 q items + 144 kv items)
// -------------------------------------------------------------------------
__global__ void point_prep_kernel(const float* __restrict__ qp_local,
                                  const float* __restrict__ kvp_local,
                                  const float* __restrict__ rot,
                                  const float* __restrict__ trans,
                                  float* __restrict__ qp_att,
                                  float* __restrict__ kp_att,
                                  float* __restrict__ vp_att) {
  const int n = blockIdx.x;
  const int t = threadIdx.x;
  float R[9], T[3];
#pragma unroll
  for (int e = 0; e < 9; ++e) R[e] = rot[n * 9 + e];
#pragma unroll
  for (int e = 0; e < 3; ++e) T[e] = trans[n * 3 + e];

  if (t < NH * PQK) {                        // q points: 48 items
    const int h = t >> 2, p = t & 3;
    float l0 = qp_local[(size_t)n * 144 + 0 * 48 + h * PQK + p];
    float l1 = qp_local[(size_t)n * 144 + 1 * 48 + h * PQK + p];
    float l2 = qp_local[(size_t)n * 144 + 2 * 48 + h * PQK + p];
#pragma unroll
    for (int d = 0; d < 3; ++d) {
      float g = R[d * 3 + 0] * l0 + R[d * 3 + 1] * l1 + R[d * 3 + 2] * l2 + T[d];
      qp_att[((size_t)n * NH + h) * 12 + p * 3 + d] = g;
    }
  } else {                                   // k/v points: 144 items
    const int idx = t - NH * PQK;
    const int h = idx / 12, c = idx % 12;
    float l0 = kvp_local[(size_t)n * 432 + 0 * 144 + h * 12 + c];
    float l1 = kvp_local[(size_t)n * 432 + 1 * 144 + h * 12 + c];
    float l2 = kvp_local[(size_t)n * 432 + 2 * 144 + h * 12 + c];
#pragma unroll
    for (int d = 0; d < 3; ++d) {
      float g = R[d * 3 + 0] * l0 + R[d * 3 + 1] * l1 + R[d * 3 + 2] * l2 + T[d];
      if (c < PQK) kp_att[((size_t)n * NH + h) * 12 + c * 3 + d] = g;
      else         vp_att[((size_t)n * NH + h) * 24 + (c - PQK) * 3 + d] = g;
    }
  }
}

// Squared norms of q/k points per (n,h).  grid = NS, block = 32.
__global__ void point_norm_kernel(const float* __restrict__ qp_att,
                                  const float* __restrict__ kp_att,
                                  float* __restrict__ qn, float* __restrict__ kn) {
  const int n = blockIdx.x, h = threadIdx.x;
  if (h >= NH) return;
  float sq = 0.f, sk = 0.f;
#pragma unroll
  for (int c = 0; c < 12; ++c) {
    float a = qp_att[((size_t)n * NH + h) * 12 + c];
    float b = kp_att[((size_t)n * NH + h) * 12 + c];
    sq += a * a; sk += b * b;
  }
  qn[(size_t)n * NH + h] = sq;
  kn[(size_t)n * NH + h] = sk;
}

// -------------------------------------------------------------------------
// logits[h][i][j]  = attn2d_w * (inputs_2d[i,j,:] @ w2d[:,h] + b2d[h])
// Flattened (i,j) is the M dimension; heads (padded 12->16) the N dimension.
// grid = NS*NS/16, block = 32.
// -------------------------------------------------------------------------
__global__ void attn2d_kernel(const float* __restrict__ in2d,
                              const float* __restrict__ w2d,
                              const float* __restrict__ b2d,
                              float* __restrict__ logits) {
  const int lane = threadIdx.x & 31;
  const int lh = lane >> 4, l16 = lane & 15;
  const size_t m0 = (size_t)blockIdx.x * 16;
  const int i = (int)(m0 / NS);
  const int j0 = (int)(m0 % NS);
  const bool hok = (l16 < NH);
  v8f acc = {};
  for (int k = 0; k < CPD; k += 4) {
    const int ka = k + 2 * lh;
    v2f a, b;
    const float* ap = in2d + (m0 + l16) * CPD + ka;
    __builtin_prefetch(ap + 16 * CPD, 0, 0);   // stream 302MB through: prefetch ahead
    a.x = ap[0]; a.y = ap[1];
    b.x = hok ? w2d[(size_t)ka * NH + l16] : 0.f;
    b.y = hok ? w2d[(size_t)(ka + 1) * NH + l16] : 0.f;
    acc = wmma_f32(a, b, acc);
  }
  if (hok) {
    const int h = l16;
    const float bb = b2d[h];
#pragma unroll
    for (int v = 0; v < 8; ++v) {
      const int j = j0 + v + 8 * lh;
      logits[((size_t)h * NS + i) * NS + j] = 0.57735026919f * (acc[v] + bb);
    }
  }
}

// -------------------------------------------------------------------------
// Adds scalar QK + point QK + mask to logits.  Point term uses
// -0.5*pw*(|q|^2 + |k|^2 - 2 q.k)  with q.k on WMMA (K=12).
// grid = (NS/16, NS/16, NH), block = 32.
// -------------------------------------------------------------------------
__global__ void qk_logits_kernel(const float* __restrict__ q_scalar,
                                 const float* __restrict__ kv,
                                 const float* __restrict__ qp_att,
                                 const float* __restrict__ kp_att,
                                 const float* __restrict__ qn,
                                 const float* __restrict__ kn,
                                 const float* __restrict__ mask,
                                 const float* __restrict__ tpw,
                                 float* __restrict__ logits) {
  const int lane = threadIdx.x & 31;
  const int lh = lane >> 4, l16 = lane & 15;
  const int i0 = blockIdx.x * 16, j0 = blockIdx.y * 16, h = blockIdx.z;
  const float sw = 0.14433756729f;  // sqrt(1/(3*SQK))
  v8f accS = {}, accP = {};
#pragma unroll
  for (int k = 0; k < SQK; k += 4) {       // scalar q.k, K=16
    const int ka = k + 2 * lh;
    v2f a, b;
    const float* qp = q_scalar + (size_t)(i0 + l16) * (NH * SQK) + h * SQK + ka;
    a.x = sw * qp[0]; a.y = sw * qp[1];
    const float* kp = kv + (size_t)(j0 + l16) * (NH * 32) + h * 32 + ka;
    b.x = kp[0]; b.y = kp[1];
    accS = wmma_f32(a, b, accS);
  }
#pragma unroll
  for (int k = 0; k < 12; k += 4) {        // point q.k, K=12
    const int ka = k + 2 * lh;
    v2f a, b;
    const float* ap = qp_att + ((size_t)(i0 + l16) * NH + h) * 12 + ka;
    a.x = ap[0]; a.y = ap[1];
    const float* bp = kp_att + ((size_t)(j0 + l16) * NH + h) * 12 + ka;
    b.x = bp[0]; b.y = bp[1];
    accP = wmma_f32(a, b, accP);
  }
  // point_weights = sqrt(1/(3*PQK*9/2)) * softplus(tpw)
  const float pw = 0.13608276348f * log1pf(expf(tpw[h]));
  const int j = j0 + l16;
  const float mj = mask[j];
  const float knj = kn[(size_t)j * NH + h];
#pragma unroll
  for (int v = 0; v < 8; ++v) {
    const int i = i0 + v + 8 * lh;
    const float d2 = qn[(size_t)i * NH + h] + knj - 2.f * accP[v];
    float val = accS[v] - 0.5f * pw * d2 - 1.0e9f * (1.f - mask[i] * mj);
    logits[((size_t)h * NS + i) * NS + j] += val;
  }
}

// Row softmax over j (length NS=768). grid = NH*NS, block = 256.
__global__ void softmax_kernel(float* __restrict__ logits) {
  __shared__ float red[256];
  float* p = logits + (size_t)blockIdx.x * NS;
  const int t = threadIdx.x;
  float vals[3];
  float m = -INFINITY;
#pragma unroll
  for (int e = 0; e < 3; ++e) { vals[e] = p[t + e * 256]; m = fmaxf(m, vals[e]); }
  red[t] = m; __syncthreads();
  for (int s = 128; s > 0; s >>= 1) { if (t < s) red[t] = fmaxf(red[t], red[t + s]); __syncthreads(); }
  m = red[0]; __syncthreads();
  float sum = 0.f;
#pragma unroll
  for (int e = 0; e < 3; ++e) { vals[e] = expf(vals[e] - m); sum += vals[e]; }
  red[t] = sum; __syncthreads();
  for (int s = 128; s > 0; s >>= 1) { if (t < s) red[t] += red[t + s]; __syncthreads(); }
  const float inv = 1.f / red[0];
#pragma unroll
  for (int e = 0; e < 3; ++e) p[t + e * 256] = vals[e] * inv;
}

// -------------------------------------------------------------------------
// attn @ v_scalar (into final_act[:,0:192]) and attn @ v_point (into rp_glob),
// sharing the attn A-fragment across 3 accumulators. grid=(NS/16, NH), blk=32
// -------------------------------------------------------------------------
__global__ void attn_v_kernel(const float* __restrict__ attn,
                              const float* __restrict__ kv,
                              const float* __restrict__ vp_att,
                              float* __restrict__ final_act,
                              float* __restrict__ rp_glob) {
  const int lane = threadIdx.x & 31;
  const int lh = lane >> 4, l16 = lane & 15;
  const int i0 = blockIdx.x * 16, h = blockIdx.y;
  const int c1 = 16 + l16;                 // second point tile column (valid < 24)
  v8f accV = {}, accP0 = {}, accP1 = {};
  for (int k = 0; k < NS; k += 4) {
    const int ka = k + 2 * lh;
    v2f a;
    const float* ap = attn + ((size_t)h * NS + (i0 + l16)) * NS + ka;
    a.x = ap[0]; a.y = ap[1];
    v2f bv, b0, b1;
    bv.x = kv[(size_t)ka * (NH * 32) + h * 32 + SQK + l16];
    bv.y = kv[(size_t)(ka + 1) * (NH * 32) + h * 32 + SQK + l16];
    b0.x = vp_att[(size_t)ka * (NH * 24) + h * 24 + l16];
    b0.y = vp_att[(size_t)(ka + 1) * (NH * 24) + h * 24 + l16];
    b1.x = (c1 < 24) ? vp_att[(size_t)ka * (NH * 24) + h * 24 + c1] : 0.f;
    b1.y = (c1 < 24) ? vp_att[(size_t)(ka + 1) * (NH * 24) + h * 24 + c1] : 0.f;
    accV = wmma_f32(a, bv, accV);
    accP0 = wmma_f32(a, b0, accP0);
    accP1 = wmma_f32(a, b1, accP1);
  }
#pragma unroll
  for (int v = 0; v < 8; ++v) {
    const int i = i0 + v + 8 * lh;
    final_act[(size_t)i * FACT_DIM + h * SV + l16] = accV[v];
    rp_glob[(size_t)i * (NH * 24) + h * 24 + l16] = accP0[v];
    if (c1 < 24) rp_glob[(size_t)i * (NH * 24) + h * 24 + c1] = accP1[v];
  }
}

// -------------------------------------------------------------------------
// attn_over_2d: out[i,h,c] = sum_j attn[h,i,j] * inputs_2d[i,j,c]
// heads are the (padded) M rows. grid = (NS, CPD/16), block = 32.
// The single-use 302MB inputs_2d stream is staged LDS-side with the CDNA5
// async copy path (GLOBAL_LOAD_ASYNC_TO_LDS_B128 + ASYNCcnt), double-buffered
// so the next 16x16 chunk's DMA overlaps the current chunk's 4 WMMAs.
// -------------------------------------------------------------------------
__global__ void attn_over2d_kernel(const float* __restrict__ attn,
                                   const float* __restrict__ in2d,
                                   float* __restrict__ final_act) {
  __shared__ float tile[2][16][16];        // 2 x 1KB chunk: [j-in-chunk][c]
  const int lane = threadIdx.x & 31;
  const int lh = lane >> 4, l16 = lane & 15;
  const int i = blockIdx.x;
  const int c0 = blockIdx.y * 16;
  const bool hok = (l16 < NH);

  // Async-copy geometry: issue t (0/1) + lane covers row (lane>>2)+8t,
  // 16 contiguous bytes at col 4*(lane&3). Generic->LDS pointers carry the
  // LDS byte offset in their low 32 bits (aperture layout), which is what
  // VDST of GLOBAL_LOAD_ASYNC_TO_LDS wants.
  const int arow = lane >> 2;              // 0..7
  const int acol = (lane & 3) * 4;         // 0,4,8,12
  const unsigned ldsBase =
      (unsigned)(uintptr_t)&tile[0][0][0] + (unsigned)(lane * 16);

  // prologue: chunk 0 -> buffer 0
  {
    const float* g0 = in2d + ((size_t)i * NS + arow) * CPD + c0 + acol;
    const float* g1 = in2d + ((size_t)i * NS + arow + 8) * CPD + c0 + acol;
    asm volatile("global_load_async_to_lds_b128 %0, %1, off"
                 :: "v"(ldsBase), "v"((unsigned long long)(uintptr_t)g0)
                 : "memory");
    asm volatile("global_load_async_to_lds_b128 %0, %1, off"
                 :: "v"(ldsBase + 512u), "v"((unsigned long long)(uintptr_t)g1)
                 : "memory");
  }

  v8f acc = {};
  for (int ch = 0; ch < NS / 16; ++ch) {
    const int buf = ch & 1;
    if (ch + 1 < NS / 16) {                // issue next chunk, then wait current
      const int k0 = (ch + 1) * 16;
      const unsigned ldsb = ldsBase + (unsigned)(((ch + 1) & 1) * 1024);
      const float* g0 = in2d + ((size_t)i * NS + k0 + arow) * CPD + c0 + acol;
      const float* g1 = in2d + ((size_t)i * NS + k0 + arow + 8) * CPD + c0 + acol;
      asm volatile("global_load_async_to_lds_b128 %0, %1, off"
                   :: "v"(ldsb), "v"((unsigned long long)(uintptr_t)g0)
                   : "memory");
      asm volatile("global_load_async_to_lds_b128 %0, %1, off"
                   :: "v"(ldsb + 512u), "v"((unsigned long long)(uintptr_t)g1)
                   : "memory");
      asm volatile("s_wait_asynccnt 0x2" ::: "memory");   // current chunk landed
    } else {
      asm volatile("s_wait_asynccnt 0x0" ::: "memory");   // last chunk landed
    }
    const int kbase = ch * 16;
#pragma unroll
    for (int kk = 0; kk < 16; kk += 4) {
      const int r = kk + 2 * lh;           // B row inside the staged chunk
      const int ka = kbase + r;            // global j for the A fragment
      v2f a, b;
      a.x = hok ? attn[((size_t)l16 * NS + i) * NS + ka] : 0.f;
      a.y = hok ? attn[((size_t)l16 * NS + i) * NS + ka + 1] : 0.f;
      b.x = tile[buf][r][l16];
      b.y = tile[buf][r + 1][l16];
      acc = wmma_f32(a, b, acc);
    }
  }
#pragma unroll
  for (int v = 0; v < 8; ++v) {
    const int hh = v + 8 * lh;
    if (hh < NH)
      final_act[(size_t)i * FACT_DIM + 576 + hh * CPD + c0 + l16] = acc[v];
  }
}

// Inverse-frame rotation of attended points + norms -> final_act[192:576].
// grid = NS, block = 96 (= H*PV).
__global__ void rp_finish_kernel(const float* __restrict__ rp_glob,
                                 const float* __restrict__ rot,
                                 const float* __restrict__ trans,
                                 float* __restrict__ final_act) {
  const int n = blockIdx.x, t = threadIdx.x;
  const int h = t >> 3, p = t & 7;
  float R[9], T[3];
#pragma unroll
  for (int e = 0; e < 9; ++e) R[e] = rot[n * 9 + e];
#pragma unroll
  for (int e = 0; e < 3; ++e) T[e] = trans[n * 3 + e];
  const float g0 = rp_glob[((size_t)n * NH + h) * 24 + p * 3 + 0] - T[0];
  const float g1 = rp_glob[((size_t)n * NH + h) * 24 + p * 3 + 1] - T[1];
  const float g2 = rp_glob[((size_t)n * NH + h) * 24 + p * 3 + 2] - T[2];
  // local[i] = sum_j R[j][i] * g[j]  (inverse rotation)
  const float lx = R[0] * g0 + R[3] * g1 + R[6] * g2;
  const float ly = R[1] * g0 + R[4] * g1 + R[7] * g2;
  const float lz = R[2] * g0 + R[5] * g1 + R[8] * g2;
  const float nrm = sqrtf(1e-8f + lx * lx + ly * ly + lz * lz);
  float* fa = final_act + (size_t)n * FACT_DIM;
  fa[192 + t] = lx;
  fa[288 + t] = ly;
  fa[384 + t] = lz;
  fa[480 + t] = nrm;
}

// -------------------------------------------------------------------------
// Workspace layout (floats). Total ~10.27M floats (~41 MB).
// -------------------------------------------------------------------------
#define OFF_Q      0u          // 768*192
#define OFF_KV     147456u     // 768*384
#define OFF_QPL    442368u     // 768*144
#define OFF_KVPL   552960u     // 768*432
#define OFF_QPA    884736u     // 768*12*12
#define OFF_KPA    995328u     // 768*12*12
#define OFF_VPA    1105920u    // 768*12*24
#define OFF_QN     1327104u    // 768*12
#define OFF_KN     1336320u    // 768*12
#define OFF_LOG    1345536u    // 12*768*768
#define OFF_RPG    8423424u    // 768*12*24
#define OFF_FACT   8644608u    // 768*2112

extern "C" void kernel_launch(void* const* d_in, const int* in_sizes, int n_in,
                              void* d_out, int out_size, void* d_ws, size_t ws_size,
                              hipStream_t stream) {
  (void)in_sizes; (void)n_in; (void)out_size; (void)ws_size;
  const float* inputs_1d = (const float*)d_in[0];
  const float* inputs_2d = (const float*)d_in[1];
  const float* mask      = (const float*)d_in[2];
  const float* rot       = (const float*)d_in[3];
  const float* trans     = (const float*)d_in[4];
  const float* wq        = (const float*)d_in[5];
  const float* bq        = (const float*)d_in[6];
  const float* wkv       = (const float*)d_in[7];
  const float* bkv       = (const float*)d_in[8];
  const float* wqp       = (const float*)d_in[9];
  const float* bqp       = (const float*)d_in[10];
  const float* wkvp      = (const float*)d_in[11];
  const float* bkvp      = (const float*)d_in[12];
  const float* w2d       = (const float*)d_in[13];
  const float* b2d       = (const float*)d_in[14];
  const float* tpw       = (const float*)d_in[15];
  const float* wout      = (const float*)d_in[16];
  const float* bout      = (const float*)d_in[17];
  float* out = (float*)d_out;
  float* ws  = (float*)d_ws;

  float* q_scalar  = ws + OFF_Q;
  float* kv        = ws + OFF_KV;
  float* qp_local  = ws + OFF_QPL;
  float* kvp_local = ws + OFF_KVPL;
  float* qp_att    = ws + OFF_QPA;
  float* kp_att    = ws + OFF_KPA;
  float* vp_att    = ws + OFF_VPA;
  float* qn        = ws + OFF_QN;
  float* kn        = ws + OFF_KN;
  float* logits    = ws + OFF_LOG;
  float* rp_glob   = ws + OFF_RPG;
  float* final_act = ws + OFF_FACT;

  const dim3 wave(32);
  // 1) linear projections of inputs_1d (K=384)
  gemm_wmma_kernel<<<dim3(192 / 16, NS / 16), wave, 0, stream>>>(inputs_1d, wq,   bq,   q_scalar,  NS, CM, 192);
  gemm_wmma_kernel<<<dim3(384 / 16, NS / 16), wave, 0, stream>>>(inputs_1d, wkv,  bkv,  kv,        NS, CM, 384);
  gemm_wmma_kernel<<<dim3(144 / 16, NS / 16), wave, 0, stream>>>(inputs_1d, wqp,  bqp,  qp_local,  NS, CM, 144);
  gemm_wmma_kernel<<<dim3(432 / 16, NS / 16), wave, 0, stream>>>(inputs_1d, wkvp, bkvp, kvp_local, NS, CM, 432);
  // 2) rotate/translate points into global frame, repack, norms
  point_prep_kernel<<<NS, 192, 0, stream>>>(qp_local, kvp_local, rot, trans, qp_att, kp_att, vp_att);
  point_norm_kernel<<<NS, 32, 0, stream>>>(qp_att, kp_att, qn, kn);
  // 3) 2d-bias projection (initializes logits)
  attn2d_kernel<<<(NS * NS) / 16, wave, 0, stream>>>(inputs_2d, w2d, b2d, logits);
  // 4) scalar + point attention terms + mask
  qk_logits_kernel<<<dim3(NS / 16, NS / 16, NH), wave, 0, stream>>>(
      q_scalar, kv, qp_att, kp_att, qn, kn, mask, tpw, logits);
  // 5) softmax over j
  softmax_kernel<<<NH * NS, 256, 0, stream>>>(logits);
  // 6) apply attention
  attn_v_kernel<<<dim3(NS / 16, NH), wave, 0, stream>>>(logits, kv, vp_att, final_act, rp_glob);
  attn_over2d_kernel<<<dim3(NS, CPD / 16), wave, 0, stream>>>(logits, inputs_2d, final_act);
  // 7) back to local frame + norms
  rp_finish_kernel<<<NS, 96, 0, stream>>>(rp_glob, rot, trans, final_act);
  // 8) output projection
  gemm_wmma_kernel<<<dim3(COUT / 16, NS / 16), wave, 0, stream>>>(final_act, wout, bout, out, NS, FACT_DIM, COUT);
}